// ChamferDistance_3762391351367
// MI455X (gfx1250) — compile-verified
//
#include <hip/hip_runtime.h>
#include <hip/hip_bf16.h>

typedef __attribute__((ext_vector_type(2))) float v2f;
typedef __attribute__((ext_vector_type(8))) float v8f;

#define CD_B 8
#define CD_N 4096
#define CD_TILES (CD_N / 16)   // 256

// ---------------------------------------------------------------------------
// Pack [N,3] f32 points into float4 (-2x, -2y, -2z, ||p||^2).
// This layout IS the WMMA B-fragment: lane-half 0 reads bytes [0,8) of a
// point (K=0,1), lane-half 1 reads bytes [8,16) (K=2,3) — a single b64 load,
// no in-loop scaling or component selects.
// ---------------------------------------------------------------------------
__global__ void cd_pack_kernel(const float* __restrict__ src,
                               float4* __restrict__ dst, int npts) {
    int i = blockIdx.x * blockDim.x + threadIdx.x;
    if (i < npts) {
        float x = src[3 * i + 0];
        float y = src[3 * i + 1];
        float z = src[3 * i + 2];
        dst[i] = make_float4(-2.0f * x, -2.0f * y, -2.0f * z,
                             x * x + y * y + z * z);
    }
}

// ---------------------------------------------------------------------------
// One wave owns a 16-row strip of P and sweeps all of Q with
// V_WMMA_F32_16X16X4_F32:  A row n = [x, y, z, 1], B col m = [-2y, ..., ||y||^2]
//   D[n,m] = -2*P[n].Q[m] + ||Q[m]||^2
// Running elementwise min over tiles, shfl_xor row-min across lanes,
// then + ||P[n]||^2 gives min_m dist(P[n], Q[m]).
// Block = 256 threads = 8 waves = 8 P-tiles.
// ---------------------------------------------------------------------------
__global__ void cd_min_kernel(const float4* __restrict__ Ppk,   // [B*N] packed
                              const float4* __restrict__ Qpk,   // [B*N] packed
                              float* __restrict__ outmin) {     // [B*N]
    const int b     = blockIdx.y;
    const int wave  = threadIdx.x >> 5;
    const int lane  = threadIdx.x & 31;
    const int half  = lane >> 4;     // 0: lanes 0-15 (K=0,1), 1: lanes 16-31 (K=2,3)
    const int li    = lane & 15;
    const int ptile = blockIdx.x * 8 + wave;

    const char* __restrict__ P = (const char*)(Ppk + (size_t)b * CD_N);
    const char* __restrict__ Q = (const char*)(Qpk + (size_t)b * CD_N);

    // --- A fragment: 16x4 f32, rows = P points, K = [x, y, z, 1] -----------
    // packed holds (-2x,-2y,-2z,w): recover coords with *(-0.5), K=3 -> 1.0
    const int prow = ptile * 16 + li;           // M = li for both lane halves
    v2f afrag = *(const v2f*)(P + (size_t)prow * 16 + (size_t)half * 8);
    afrag.x *= -0.5f;
    afrag.y = half ? 1.0f : (afrag.y * -0.5f);

    const v8f czero = {};
    v8f mn;
#pragma unroll
    for (int i = 0; i < 8; ++i) mn[i] = 3.0e38f;

    // per-lane B-fragment stream: one b64 per tile, stride 256 bytes
    const char* __restrict__ qlane = Q + (size_t)li * 16 + (size_t)half * 8;

    // --- sweep all Q tiles --------------------------------------------------
    for (int qt = 0; qt < CD_TILES; ++qt) {
        const v2f bfrag = *(const v2f*)(qlane + (size_t)qt * 256);

        // D = A x B + 0  ->  D[n,m] = -2*P[n].Q[m] + ||Q[m]||^2
        v8f d = __builtin_amdgcn_wmma_f32_16x16x4_f32(
            /*neg_a=*/false, afrag, /*neg_b=*/false, bfrag,
            /*c_mod=*/(short)0, czero, /*reuse_a=*/false, /*reuse_b=*/false);

#pragma unroll
        for (int i = 0; i < 8; ++i) mn[i] = fminf(mn[i], d[i]);
    }

    // --- row-min across the 16 lanes of each half (columns N) --------------
#pragma unroll
    for (int off = 1; off < 16; off <<= 1) {
#pragma unroll
        for (int i = 0; i < 8; ++i)
            mn[i] = fminf(mn[i], __shfl_xor(mn[i], off, 32));
    }

    // lane 0 holds row-mins for M=0..7; lane 16 for M=8..15 (D-matrix layout)
    if (li == 0) {
        const int rowbase = ptile * 16 + half * 8;
        float* o = outmin + (size_t)b * CD_N;
#pragma unroll
        for (int v = 0; v < 8; ++v) {
            const int r = rowbase + v;
            const float pw = *(const float*)(P + (size_t)r * 16 + 12); // ||P[n]||^2
            o[r] = mn[v] + pw;
        }
    }
}

// ---------------------------------------------------------------------------
// out[b] = mean_n(minx[b,n]) + mean_m(miny[b,m])
// ---------------------------------------------------------------------------
__global__ void cd_finalize_kernel(const float* __restrict__ minx,
                                   const float* __restrict__ miny,
                                   float* __restrict__ out) {
    const int b = blockIdx.x;
    const int t = threadIdx.x;
    __shared__ float sm[256];
    float s = 0.0f;
    for (int i = t; i < CD_N; i += 256)
        s += minx[(size_t)b * CD_N + i] + miny[(size_t)b * CD_N + i];
    sm[t] = s;
    __syncthreads();
    for (int off = 128; off > 0; off >>= 1) {
        if (t < off) sm[t] += sm[t + off];
        __syncthreads();
    }
    if (t == 0) out[b] = sm[0] * (1.0f / (float)CD_N);
}

extern "C" void kernel_launch(void* const* d_in, const int* in_sizes, int n_in,
                              void* d_out, int out_size, void* d_ws, size_t ws_size,
                              hipStream_t stream) {
    const float* x = (const float*)d_in[0];   // [8, 4096, 3] f32
    const float* y = (const float*)d_in[1];   // [8, 4096, 3] f32
    float* out = (float*)d_out;               // [8] f32

    const int npts = CD_B * CD_N;             // 32768 per cloud

    // workspace layout
    char* ws = (char*)d_ws;
    float4* xp   = (float4*)(ws);                                   // 512 KB
    float4* yp   = (float4*)(ws + (size_t)npts * sizeof(float4));   // 512 KB
    float*  minx = (float*)(ws + (size_t)2 * npts * sizeof(float4));          // 128 KB
    float*  miny = (float*)(ws + (size_t)2 * npts * sizeof(float4)
                               + (size_t)npts * sizeof(float));               // 128 KB

    // pack points + squared norms (pre-scaled into B-fragment layout)
    cd_pack_kernel<<<dim3((npts + 255) / 256), dim3(256), 0, stream>>>(x, xp, npts);
    cd_pack_kernel<<<dim3((npts + 255) / 256), dim3(256), 0, stream>>>(y, yp, npts);

    // x -> nearest y   (grid: 32 blocks of 8 waves = 256 P-tiles, per batch)
    cd_min_kernel<<<dim3(CD_TILES / 8, CD_B), dim3(256), 0, stream>>>(xp, yp, minx);
    // y -> nearest x
    cd_min_kernel<<<dim3(CD_TILES / 8, CD_B), dim3(256), 0, stream>>>(yp, xp, miny);

    // means
    cd_finalize_kernel<<<dim3(CD_B), dim3(256), 0, stream>>>(minx, miny, out);
}